// LogHarmonicLowering_59433757442130
// MI455X (gfx1250) — compile-verified
//
#include <hip/hip_runtime.h>

// LogHarmonicLowering for MI455X (gfx1250).
// Bandwidth-bound 2-tap frequency-axis interpolation, 5 harmonics.
//   in : x   (8, 32, 256, 512) f32
//   out:     (8, 160, 256, 512) f32   out[b, k*32+c, f, t]
//
// Roofline: 671 MB out + 134 MB in => ~35 us at 23.3 TB/s; AI ~ 0.1 FLOP/B.
// Pure bandwidth problem -> no WMMA. CDNA5 paths used:
//   * global_load_async_to_lds_b128 (ASYNCcnt) to stage a (256 x 64) tile once
//     and serve all 5 harmonics (10 taps) from LDS,
//   * non-temporal b128 stores so the streamed output never evicts the 134 MB
//     input from the 192 MB L2.
//
// Weight algebra: f is an integer, so w1 = frac(f+shift) = frac(shift) is a
// per-harmonic CONSTANT and y0 = f + floor(shift) is an integer add. All
// transcendental/float floor work is hoisted out of the inner loop.

#define FK   5
#define NB   8
#define NC   32
#define FR   256
#define NT   512
#define TT   64            // t-tile width (floats) -> LDS tile = 256*64*4 = 64 KB

typedef __attribute__((ext_vector_type(4))) float v4f;

__global__ __launch_bounds__(256, 1)
void loghl_kernel(const float* __restrict__ x, float* __restrict__ out)
{
    __shared__ float tile[FR * TT];            // 64 KB of the 320 KB WGP LDS

    const int tid = threadIdx.x;               // 0..255 (8 wave32s)
    const int bc  = blockIdx.x;                // 0..255  -> (b,c)
    const int t0  = blockIdx.y * TT;           // t-tile start
    const int b   = bc >> 5;
    const int c   = bc & 31;

    // ---------------- async stage: x[b,c, 0:256, t0:t0+64] -> LDS ----------------
    const float*   src      = x + ((size_t)bc * FR) * NT + t0;
    const unsigned lds_base = (unsigned)(size_t)(&tile[0]);   // flat->LDS offset (addr[31:0])

#pragma unroll
    for (int i = 0; i < (FR * TT) / (256 * 4); ++i) {         // 16 rounds of b128 per thread
        const int li  = i * 256 + tid;                        // which float4 of the tile
        const int row = li >> 4;                              // 16 float4 per row
        const int c4  = (li & 15) << 2;                       // starting column (floats)
        const unsigned lds_addr = lds_base + (unsigned)((row * TT + c4) * 4);
        const float*   gp       = src + (size_t)row * NT + c4;
        asm volatile("global_load_async_to_lds_b128 %0, %1, off"
                     :: "v"(lds_addr), "v"(gp)
                     : "memory");
    }
    asm volatile("s_wait_asynccnt 0x0" ::: "memory");
    __syncthreads();

    // ---------------- compute: 5 harmonics x 256 rows from LDS ----------------
    // shift_px[k] = 12*log2(k+1) * 256/255 ; split into int displacement + frac.
    const int   dint [FK] = {0, 12, 19, 24, 27};
    const float wfrac[FK] = {0.0f, 0.0470588f, 0.0941372f, 0.0941177f, 0.9724045f};

    const int cg = (tid & 15) << 2;     // column group: 4 floats
    const int fb = tid >> 4;            // 0..15 : f-phase within a 16-row pass

    // fb*NT folded into the base; it*16*NT folds into the 24-bit store offset.
    const size_t obase0 = (((size_t)b * (FK * NC) + c) * FR + fb) * NT + t0 + cg;

#pragma unroll
    for (int k = 0; k < FK; ++k) {
        const int   d   = dint[k];
        const float w1k = wfrac[k];
        const float w0k = 1.0f - w1k;
        float* obase = out + obase0 + (size_t)(k * NC) * FR * NT;
#pragma unroll
        for (int it = 0; it < FR / 16; ++it) {
            const int f  = it * 16 + fb;
            const int y0 = f + d;                       // always >= 0 (d >= 0)
            // boundary masks: compile-time true/false for all but one it-block per k
            const float wv0 = (y0 <= FR - 1) ? w0k : 0.0f;
            const float wv1 = (y0 <= FR - 2) ? w1k : 0.0f;
            const int   y0c = min(y0,     FR - 1);
            const int   y1c = min(y0 + 1, FR - 1);

            const v4f a  = *(const v4f*)&tile[y0c * TT + cg];  // ds_load_b128
            const v4f bb = *(const v4f*)&tile[y1c * TT + cg];  // ds_load_b128

            v4f r;
            r.x = wv0 * a.x + wv1 * bb.x;
            r.y = wv0 * a.y + wv1 * bb.y;
            r.z = wv0 * a.z + wv1 * bb.z;
            r.w = wv0 * a.w + wv1 * bb.w;

            __builtin_nontemporal_store(r, (v4f*)(obase + (size_t)(it * 16) * NT));
        }
    }
}

extern "C" void kernel_launch(void* const* d_in, const int* in_sizes, int n_in,
                              void* d_out, int out_size, void* d_ws, size_t ws_size,
                              hipStream_t stream)
{
    const float* x   = (const float*)d_in[0];
    float*       out = (float*)d_out;

    dim3 grid(NB * NC, NT / TT);   // (256, 8) blocks
    dim3 block(256);               // 8 wave32s per WGP-resident workgroup
    hipLaunchKernelGGL(loghl_kernel, grid, block, 0, stream, x, out);
}